// GNN_47923245088958
// MI455X (gfx1250) — compile-verified
//
#include <hip/hip_runtime.h>
#include <math.h>

typedef __attribute__((ext_vector_type(2))) float v2f;
typedef __attribute__((ext_vector_type(8))) float v8f;

// ---------------------------------------------------------------------------
// Graph preprocessing: degree (incl. self loop), d^{-1/2}, per-edge GCN norm
// ---------------------------------------------------------------------------
__global__ void gcn_deg_init(float* __restrict__ deg, int N) {
    int i = blockIdx.x * blockDim.x + threadIdx.x;
    if (i < N) deg[i] = 1.0f;                 // self-loop contributes 1
}

__global__ void gcn_deg_count(const int* __restrict__ col, float* __restrict__ deg, int E) {
    int e = blockIdx.x * blockDim.x + threadIdx.x;
    if (e < E) atomicAdd(&deg[col[e]], 1.0f);
}

__global__ void gcn_dinv(const float* __restrict__ deg, float* __restrict__ dinv, int N) {
    int i = blockIdx.x * blockDim.x + threadIdx.x;
    if (i < N) dinv[i] = rsqrtf(deg[i]);      // deg >= 1 always (self loops)
}

__global__ void gcn_edge_norm(const int* __restrict__ row, const int* __restrict__ col,
                              const float* __restrict__ dinv, float* __restrict__ norm, int E) {
    int e = blockIdx.x * blockDim.x + threadIdx.x;
    if (e < E) norm[e] = dinv[row[e]] * dinv[col[e]];
}

__global__ void gcn_zero(float* __restrict__ p, int n) {
    int i = blockIdx.x * blockDim.x + threadIdx.x;
    int base = i * 4;
    if (base + 3 < n) {
        *(float4*)(p + base) = make_float4(0.f, 0.f, 0.f, 0.f);
    } else {
        for (int k = base; k < n; ++k) p[k] = 0.0f;
    }
}

// ---------------------------------------------------------------------------
// Dense GEMM  H[N,d_out] = X[N,d_in] @ W[d_in,d_out]  in fp32 via
// V_WMMA_F32_16X16X4_F32.  One wave32 owns one 16x16 output tile.
//   A (16x4): lane m=lane&15; VGPR0/1 = K = k + (lane>=16 ? 2:0) + {0,1}
//   B (4x16): lane n=lane&15; same K pattern, column n of W
//   C/D (16x16): VGPR r holds M = r + (lane>=16 ? 8:0), N = lane&15
// ---------------------------------------------------------------------------
__global__ void gcn_gemm_wmma(const float* __restrict__ X, const float* __restrict__ W,
                              float* __restrict__ H, int N, int d_in, int d_out) {
    const int wave = threadIdx.x >> 5;
    const int lane = threadIdx.x & 31;
    const int tiles_n = d_out >> 4;
    const int tiles_m = (N + 15) >> 4;
    const int tile_id = blockIdx.x * (blockDim.x >> 5) + wave;
    if (tile_id >= tiles_m * tiles_n) return;          // wave-uniform: EXEC all-1s at WMMA
    const int tile_m = tile_id / tiles_n;
    const int tile_n = tile_id % tiles_n;

    const int mn    = lane & 15;                       // row for A, col for B
    const int khalf = (lane >> 4) << 1;                // 0 or 2

    const float* __restrict__ xrow = X + (size_t)(tile_m * 16 + mn) * d_in;
    const float* __restrict__ wcol = W + tile_n * 16 + mn;

    v8f acc = {};
    for (int k = 0; k < d_in; k += 4) {
        v2f a = *(const v2f*)(xrow + k + khalf);       // contiguous K pair
        v2f b;
        b.x = wcol[(size_t)(k + khalf)     * d_out];
        b.y = wcol[(size_t)(k + khalf + 1) * d_out];
        acc = __builtin_amdgcn_wmma_f32_16x16x4_f32(
                  /*neg_a=*/false, a, /*neg_b=*/false, b,
                  /*c_mod=*/(short)0, acc, /*reuse_a=*/false, /*reuse_b=*/false);
    }

    const int m_base = tile_m * 16 + ((lane >> 4) << 3);
    float* __restrict__ hp = H + (size_t)m_base * d_out + tile_n * 16 + mn;
#pragma unroll
    for (int r = 0; r < 8; ++r) hp[(size_t)r * d_out] = acc[r];
}

// ---------------------------------------------------------------------------
// Output layer GEMV: H[N] = X[N,512] @ W[512]; one wave per node
// ---------------------------------------------------------------------------
__global__ void gcn_gemv512(const float* __restrict__ X, const float* __restrict__ W,
                            float* __restrict__ H, int N) {
    const int node = blockIdx.x * (blockDim.x >> 5) + (threadIdx.x >> 5);
    const int lane = threadIdx.x & 31;
    if (node >= N) return;
    const float* __restrict__ xp = X + (size_t)node * 512;
    float s = 0.0f;
#pragma unroll 4
    for (int k = lane; k < 512; k += 32) s += xp[k] * W[k];
#pragma unroll
    for (int off = 16; off > 0; off >>= 1) s += __shfl_xor(s, off, 32);
    if (lane == 0) H[node] = s;
}

// ---------------------------------------------------------------------------
// Edge scatter: agg[col[e], :] += norm[e] * H[row[e], :]
// One thread per (edge, 4-feature chunk); d is a power of two (shift given).
// ---------------------------------------------------------------------------
__global__ void gcn_scatter_v4(const int* __restrict__ row, const int* __restrict__ col,
                               const float* __restrict__ norm, const float* __restrict__ H,
                               float* __restrict__ agg, int E, int d, int vshift) {
    const int gid  = blockIdx.x * blockDim.x + threadIdx.x;
    const int vecs = d >> 2;
    if (gid >= E * vecs) return;
    const int e = gid >> vshift;          // vshift = log2(vecs)
    const int v = gid & (vecs - 1);
    const float  nr = norm[e];
    const float4 hv = *((const float4*)(H + (size_t)row[e] * d) + v);
    float* __restrict__ ap = agg + (size_t)col[e] * d + (v << 2);
    atomicAdd(ap + 0, nr * hv.x);
    atomicAdd(ap + 1, nr * hv.y);
    atomicAdd(ap + 2, nr * hv.z);
    atomicAdd(ap + 3, nr * hv.w);
}

__global__ void gcn_scatter_s1(const int* __restrict__ row, const int* __restrict__ col,
                               const float* __restrict__ norm, const float* __restrict__ H,
                               float* __restrict__ agg, int E) {
    const int e = blockIdx.x * blockDim.x + threadIdx.x;
    if (e < E) atomicAdd(&agg[col[e]], norm[e] * H[row[e]]);
}

// ---------------------------------------------------------------------------
// Finalize: add self-loop term dinv[i]^2*H[i,:], (mean), +bias, activation
// mode 0: mean + relu        mode 1: add + sigmoid
// ---------------------------------------------------------------------------
__global__ void gcn_finalize(const float* __restrict__ H, const float* __restrict__ agg,
                             const float* __restrict__ dinv, const float* __restrict__ cnt,
                             const float* __restrict__ bias, float* __restrict__ out,
                             int N, int d, int mode) {
    const int gid = blockIdx.x * blockDim.x + threadIdx.x;
    if (gid >= N * d) return;
    const int i = gid / d;
    const int j = gid - i * d;
    const float di = dinv[i];
    float v = agg[gid] + di * di * H[gid];
    if (mode == 0) {
        v = v / cnt[i] + bias[j];
        out[gid] = v > 0.0f ? v : 0.0f;
    } else {
        v += bias[j];
        out[gid] = 1.0f / (1.0f + expf(-v));
    }
}

// ---------------------------------------------------------------------------
static inline int cdiv(int a, int b) { return (a + b - 1) / b; }

extern "C" void kernel_launch(void* const* d_in, const int* in_sizes, int n_in,
                              void* d_out, int out_size, void* d_ws, size_t ws_size,
                              hipStream_t stream) {
    const float* x   = (const float*)d_in[0];
    const int*   ei  = (const int*)d_in[1];
    const float* Wm[5] = { (const float*)d_in[2], (const float*)d_in[4], (const float*)d_in[6],
                           (const float*)d_in[8], (const float*)d_in[10] };
    const float* Bv[5] = { (const float*)d_in[3], (const float*)d_in[5], (const float*)d_in[7],
                           (const float*)d_in[9], (const float*)d_in[11] };

    const int N = in_sizes[0] / 16;
    const int E = in_sizes[1] / 2;
    const int* row = ei;           // edge_index[0] = src
    const int* col = ei + E;       // edge_index[1] = dst

    // workspace layout (floats)
    float* deg  = (float*)d_ws;              // N   (== cnt, since deg >= 1)
    float* dinv = deg + N;                   // N
    float* norm = dinv + N;                  // E
    float* buf0 = norm + E;                  // N*512  activations (X)
    float* buf1 = buf0 + (size_t)N * 512;    // N*512  GEMM output (H)
    float* buf2 = buf1 + (size_t)N * 512;    // N*512  aggregation (AGG)

    // ---- preprocessing -----------------------------------------------------
    gcn_deg_init <<<cdiv(N, 256), 256, 0, stream>>>(deg, N);
    gcn_deg_count<<<cdiv(E, 256), 256, 0, stream>>>(col, deg, E);
    gcn_dinv     <<<cdiv(N, 256), 256, 0, stream>>>(deg, dinv, N);
    gcn_edge_norm<<<cdiv(E, 256), 256, 0, stream>>>(row, col, dinv, norm, E);

    // ---- 5 GCN layers ------------------------------------------------------
    const int dims[6] = {16, 64, 128, 256, 512, 1};
    const float* X = x;
    for (int l = 0; l < 5; ++l) {
        const int din  = dims[l];
        const int dout = dims[l + 1];

        if (dout >= 16) {
            const int tiles = ((N + 15) / 16) * (dout / 16);
            gcn_gemm_wmma<<<cdiv(tiles, 4), 128, 0, stream>>>(X, Wm[l], buf1, N, din, dout);
        } else {
            gcn_gemv512<<<cdiv(N, 4), 128, 0, stream>>>(X, Wm[l], buf1, N);
        }

        const int na = N * dout;
        gcn_zero<<<cdiv(cdiv(na, 4), 256), 256, 0, stream>>>(buf2, na);

        if (dout >= 4) {
            const int vecs = dout >> 2;
            int vshift = 0; while ((1 << vshift) < vecs) ++vshift;
            gcn_scatter_v4<<<cdiv(E * vecs, 256), 256, 0, stream>>>(row, col, norm, buf1, buf2,
                                                                    E, dout, vshift);
        } else {
            gcn_scatter_s1<<<cdiv(E, 256), 256, 0, stream>>>(row, col, norm, buf1, buf2, E);
        }

        float* outp = (l == 4) ? (float*)d_out : buf0;
        gcn_finalize<<<cdiv(na, 256), 256, 0, stream>>>(buf1, buf2, dinv, deg, Bv[l], outp,
                                                        N, dout, (l == 4) ? 1 : 0);
        X = buf0;
    }
}